// DetectionPostProcess_45423574122955
// MI455X (gfx1250) — compile-verified
//
#include <hip/hip_runtime.h>
#include <hip/hip_bf16.h>
#include <stdint.h>

// ---------------------------------------------------------------------------
// Detection post-process for gfx1250 (MI455X), compile-only tuning.
//  1a: device-wide score histogram (grid 240x5x4) -> global hist atomics
//  1b: cutoff scan per (image,level) + candidate array padding init
//  1c: device-wide slot claim + inline box decode
//  2 : per-image bitonic sort (64 KB dynamic LDS)
//  3 : TDM-staged LDS-resident chunked greedy NMS + compaction
// ---------------------------------------------------------------------------

#define NUM_C      80
#define TOPK_LVL   1000
#define NUM_LVL    5
#define KTOT       (TOPK_LVL * NUM_LVL)      // 5000 candidates / image
#define DET        300
#define BATCH      4
#define SCORE_THR  0.05f
#define LOGIT_THR  (-2.9444389791664403f)    // log(0.05/0.95)
#define NMS_THR    0.5f
#define NEG_SCORE  (-1e9f)
#define EPB        16384                      // elements per block, stage 1
#define CHUNKS_MAX 240                        // ceil(49152*80 / EPB)

typedef unsigned int u32x4 __attribute__((ext_vector_type(4)));
typedef int          i32x8 __attribute__((ext_vector_type(8)));
typedef int          i32x4 __attribute__((ext_vector_type(4)));

__constant__ int c_lvl_sizes[NUM_LVL] = {49152, 12288, 3072, 768, 192};

// ---------------------------------------------------------------------------
// Tensor Data Mover: 1-D load of `nelem` DWORDs from global -> LDS.
// D# packing per CDNA5 ISA.
// Compile-time probes: if the TDM builtins are unavailable we emit #warning
// so the absence is visible in compiler stderr (compile-only feedback loop).
// ---------------------------------------------------------------------------
__device__ __forceinline__ void tdm_load_dwords_1d(void* lds_dst,
                                                   const void* gsrc,
                                                   unsigned nelem) {
#if __has_builtin(__builtin_amdgcn_tensor_load_to_lds)
  unsigned lds_addr = (unsigned)(uintptr_t)lds_dst;      // LDS byte offset
  unsigned long long ga = (unsigned long long)(uintptr_t)gsrc;

  u32x4 g0;
  g0.x = 1u;                                             // count=1, user D#
  g0.y = lds_addr;                                       // lds_addr [63:32]
  g0.z = (unsigned)(ga & 0xFFFFFFFFull);                 // global_addr lo
  g0.w = (unsigned)((ga >> 32) & 0x01FFFFFFull)          // global_addr[56:32]
       | (2u << 30);                                     // type = 2 ("image")

  i32x8 g1;
  g1[0] = (int)(2u << 16);                               // data_size=2 -> 4B
  g1[1] = (int)((nelem & 0xFFFFu) << 16);                // tensor_dim0[15:0]
  g1[2] = (int)(((nelem >> 16) & 0xFFFFu)                // tensor_dim0[31:16]
       | (1u << 16));                                    // tensor_dim1 = 1
  g1[3] = (int)((nelem & 0xFFFFu) << 16);                // tile_dim0
  g1[4] = 0;                                             // tile_dim1/2 unused
  g1[5] = (int)nelem;                                    // tensor_dim0_stride
  g1[6] = 0;
  g1[7] = 0;

  i32x4 z4 = {0, 0, 0, 0};
#if defined(__clang_major__) && (__clang_major__ >= 23)
#warning "TDM path: 6-arg __builtin_amdgcn_tensor_load_to_lds (clang>=23)"
  i32x8 z8 = {0, 0, 0, 0, 0, 0, 0, 0};
  __builtin_amdgcn_tensor_load_to_lds(g0, g1, z4, z4, z8, 0);
#else
#warning "TDM path: 5-arg __builtin_amdgcn_tensor_load_to_lds (clang<=22)"
  __builtin_amdgcn_tensor_load_to_lds(g0, g1, z4, z4, 0);
#endif
#else
#warning "TDM builtin NOT available -- falling back to per-lane LDS copy"
  const unsigned* s = (const unsigned*)gsrc;
  unsigned* d = (unsigned*)lds_dst;
  for (unsigned i = threadIdx.x & 31u; i < nelem; i += 32u) d[i] = s[i];
#endif
}

__device__ __forceinline__ void tdm_wait_all() {
#if __has_builtin(__builtin_amdgcn_s_wait_tensorcnt)
  __builtin_amdgcn_s_wait_tensorcnt(0);
#else
#warning "s_wait_tensorcnt builtin NOT available"
#endif
}

__device__ __forceinline__ float sigmoidf_fast(float x) {
  return 1.0f / (1.0f + __expf(-x));
}

// ---------------------------------------------------------------------------
// Stage 1a: device-wide histogram. grid = (CHUNKS_MAX, 5, B), block = 256.
// ---------------------------------------------------------------------------
__global__ __launch_bounds__(256)
void hist_kernel(const float* __restrict__ f0, const float* __restrict__ f1,
                 const float* __restrict__ f2, const float* __restrict__ f3,
                 const float* __restrict__ f4,
                 unsigned* __restrict__ hist) {
  const int lvl = blockIdx.y;
  const int b   = blockIdx.z;
  const int N   = c_lvl_sizes[lvl];
  const int total = N * NUM_C;
  const int start = blockIdx.x * EPB;
  if (start >= total) return;
  const int stop = min(start + EPB, total);

  const float* flist[NUM_LVL] = {f0, f1, f2, f3, f4};
  const float* __restrict__ preds = flist[lvl] + (size_t)b * (size_t)N * 84;
  unsigned* __restrict__ ghist = hist + (b * NUM_LVL + lvl) * 1024;

  __shared__ unsigned lhist[1024];
  for (int i = threadIdx.x; i < 1024; i += blockDim.x) lhist[i] = 0u;
  __syncthreads();

  for (int i = start + threadIdx.x; i < stop; i += blockDim.x) {
    const int n = i / NUM_C;
    const int c = i - n * NUM_C;
    __builtin_prefetch(&preds[n * 84 + 4 + c + blockDim.x], 0, 0);
    const float x = preds[n * 84 + 4 + c];
    if (x > LOGIT_THR) {
      const float s = sigmoidf_fast(x);
      int bkt = (int)(s * 1024.0f);
      bkt = bkt > 1023 ? 1023 : bkt;
      atomicAdd(&lhist[bkt], 1u);
    }
  }
  __syncthreads();

  for (int i = threadIdx.x; i < 1024; i += blockDim.x) {
    const unsigned v = lhist[i];
    if (v) atomicAdd(&ghist[i], v);
  }
}

// ---------------------------------------------------------------------------
// Stage 1b: cutoff per (image,level) + pad candidate arrays with NEG/zero.
// ---------------------------------------------------------------------------
__global__ __launch_bounds__(256)
void cutoff_kernel(const unsigned* __restrict__ hist,
                   int* __restrict__ cutv, int* __restrict__ budgetv,
                   int* __restrict__ sel_cnt, int* __restrict__ cut_cnt,
                   float* __restrict__ cand_score,
                   int*   __restrict__ cand_label,
                   float* __restrict__ cand_box) {
  const int g = blockIdx.x * blockDim.x + threadIdx.x;

  if (g < BATCH * NUM_LVL) {
    const unsigned* h = hist + g * 1024;
    unsigned cum = 0, above = 0;
    int cut = 0;
    for (int bkt = 1023; bkt >= 0; --bkt) {
      const unsigned v = h[bkt];
      if (cum + v >= (unsigned)TOPK_LVL || bkt == 0) { cut = bkt; above = cum; break; }
      cum += v;
    }
    cutv[g] = cut;
    budgetv[g] = TOPK_LVL - (int)above;
    sel_cnt[g] = 0;
    cut_cnt[g] = 0;
  }

  if (g < BATCH * KTOT) {
    cand_score[g] = NEG_SCORE;
    cand_label[g] = 0;
    ((float4*)cand_box)[g] = make_float4(0.f, 0.f, 0.f, 0.f);
  }
}

// ---------------------------------------------------------------------------
// Stage 1c: claim slots + inline decode. Same grid as 1a.
// ---------------------------------------------------------------------------
__global__ __launch_bounds__(256)
void select_decode_kernel(const float* __restrict__ f0, const float* __restrict__ f1,
                          const float* __restrict__ f2, const float* __restrict__ f3,
                          const float* __restrict__ f4,
                          const int* __restrict__ image_shapes,
                          const int* __restrict__ cutv, const int* __restrict__ budgetv,
                          int* __restrict__ sel_cnt, int* __restrict__ cut_cnt,
                          float* __restrict__ cand_score,
                          int*   __restrict__ cand_label,
                          float* __restrict__ cand_box) {
  const int lvl = blockIdx.y;
  const int b   = blockIdx.z;
  const int N   = c_lvl_sizes[lvl];
  const int total = N * NUM_C;
  const int start = blockIdx.x * EPB;
  if (start >= total) return;
  const int stop = min(start + EPB, total);

  const float* flist[NUM_LVL] = {f0, f1, f2, f3, f4};
  const float* __restrict__ preds = flist[lvl] + (size_t)b * (size_t)N * 84;

  const int g = b * NUM_LVL + lvl;
  const int cut = cutv[g];
  const int budget = budgetv[g];
  const int base = b * KTOT + lvl * TOPK_LVL;

  const int ih = image_shapes[b * 2 + 0];
  const int iw = image_shapes[b * 2 + 1];
  const float scl[4] = {(float)iw, (float)ih, (float)iw, (float)ih};

  for (int i = start + threadIdx.x; i < stop; i += blockDim.x) {
    const int n = i / NUM_C;
    const int c = i - n * NUM_C;
    const float x = preds[n * 84 + 4 + c];
    if (x > LOGIT_THR) {
      const float s = sigmoidf_fast(x);
      int bkt = (int)(s * 1024.0f);
      bkt = bkt > 1023 ? 1023 : bkt;
      if (bkt >= cut) {
        bool take = bkt > cut;
        if (!take) {
          const int t = atomicAdd(&cut_cnt[g], 1);
          take = t < budget;
        }
        if (take) {
          const int p = atomicAdd(&sel_cnt[g], 1);   // p < TOPK_LVL guaranteed
          float bx[4];
#pragma unroll
          for (int d = 0; d < 4; ++d) {
            float v = sigmoidf_fast(preds[n * 84 + d]) * scl[d];
            bx[d] = fminf(fmaxf(v, 0.0f), scl[d]);
          }
          cand_score[base + p] = s;
          cand_label[base + p] = c;
          ((float4*)cand_box)[base + p] = make_float4(bx[0], bx[1], bx[2], bx[3]);
        }
      }
    }
  }
}

// ---------------------------------------------------------------------------
// Stage 2: per-image bitonic sort (descending), 8192 keys in 64 KB LDS.
// ---------------------------------------------------------------------------
__global__ __launch_bounds__(1024)
void sort_candidates_kernel(const float* __restrict__ cand_score,
                            const int*   __restrict__ cand_label,
                            const float* __restrict__ cand_box,
                            float* __restrict__ sort_score,
                            int*   __restrict__ sort_label,
                            float* __restrict__ sort_box) {
  extern __shared__ unsigned long long keys[];
  const int M = 8192;
  const int b = blockIdx.x;
  const int tid = threadIdx.x;
  const int bs = blockDim.x;
  const int bK = b * KTOT;

  for (int i = tid; i < M; i += bs) {
    unsigned long long key = 0ull;               // pad: sorts last (desc)
    if (i < KTOT) {
      const float s = cand_score[bK + i];
      unsigned u = __float_as_uint(s);
      u = (u & 0x80000000u) ? ~u : (u | 0x80000000u);
      key = ((unsigned long long)u << 32) | (unsigned)i;
    }
    keys[i] = key;
  }
  __syncthreads();

  for (int k = 2; k <= M; k <<= 1) {
    for (int j = k >> 1; j > 0; j >>= 1) {
      for (int i = tid; i < M; i += bs) {
        const int ixj = i ^ j;
        if (ixj > i) {
          const bool desc_seg = ((i & k) == 0);
          const unsigned long long a = keys[i];
          const unsigned long long c = keys[ixj];
          if (desc_seg ? (a < c) : (a > c)) { keys[i] = c; keys[ixj] = a; }
        }
      }
      __syncthreads();
    }
  }

  for (int i = tid; i < KTOT; i += bs) {
    const int idx = (int)(keys[i] & 0xFFFFFFFFull);
    sort_score[bK + i] = cand_score[bK + idx];
    sort_label[bK + i] = cand_label[bK + idx];
    ((float4*)sort_box)[bK + i] = ((const float4*)cand_box)[bK + idx];
  }
}

// ---------------------------------------------------------------------------
// Stage 3: chunked greedy class-aware NMS, candidates LDS-resident via TDM.
// Exact sequential-greedy semantics with ~2 barriers per 32-candidate chunk.
// ---------------------------------------------------------------------------
__global__ __launch_bounds__(1024)
void nms_compact_kernel(const float* __restrict__ sort_score,
                        const int*   __restrict__ sort_label,
                        const float* __restrict__ sort_box,
                        float* __restrict__ out) {
  extern __shared__ char smem[];
  float4*        box = (float4*)smem;                                 // K*16 B
  int*           lab = (int*)(smem + KTOT * 16);                      // K*4 B
  unsigned char* sup = (unsigned char*)(smem + KTOT * 16 + KTOT * 4); // K B

  __shared__ float4 s_kbox[32];
  __shared__ int    s_klab[32];
  __shared__ int    s_nkept;

  const int b = blockIdx.x;
  const int tid = threadIdx.x;
  const int bs = blockDim.x;
  const int bK = b * KTOT;

  // Zero this image's output region (reference pads with zeros).
  for (int i = tid; i < DET * 6; i += bs) out[b * DET * 6 + i] = 0.0f;

  // Stage candidates into LDS via the Tensor Data Mover (wave 0 issues).
  if ((tid >> 5) == 0) {
    tdm_load_dwords_1d(box, sort_box + (size_t)bK * 4, KTOT * 4);
    tdm_load_dwords_1d(lab, sort_label + bK, KTOT);
    tdm_wait_all();
  }
  __syncthreads();

  // Invalid candidates (score <= thr) start suppressed.
  for (int i = tid; i < KTOT; i += bs)
    sup[i] = (sort_score[bK + i] > SCORE_THR) ? 0 : 1;
  __syncthreads();

  const int nChunks = (KTOT + 31) / 32;
  for (int c = 0; c < nChunks; ++c) {
    const int start = c * 32;
    const int lim = min(32, KTOT - start);

    // Phase 1: wave 0 resolves intra-chunk suppression in greedy order.
    if (tid < 32) {
      volatile unsigned char* vsup = sup;      // wave-internal DS ordering
      float4 bj; int lj = -1; float areaj = 0.f;
      if (tid < lim) {
        bj = box[start + tid];
        lj = lab[start + tid];
        areaj = (bj.z - bj.x) * (bj.w - bj.y);
      }
      for (int i = 0; i < lim; ++i) {
        const bool kept_i = (vsup[start + i] == 0);
        if (kept_i && tid > i && tid < lim && vsup[start + tid] == 0 &&
            lab[start + i] == lj) {
          const float4 bi = box[start + i];
          const float iw = fminf(bi.z, bj.z) - fmaxf(bi.x, bj.x);
          const float ih = fminf(bi.w, bj.w) - fmaxf(bi.y, bj.y);
          const float inter = fmaxf(iw, 0.0f) * fmaxf(ih, 0.0f);
          const float area_i = (bi.z - bi.x) * (bi.w - bi.y);
          const float uni = fmaxf(area_i + areaj - inter, 1e-9f);
          if (inter > NMS_THR * uni) vsup[start + tid] = 1;
        }
      }
      // Build the kept list for this chunk.
      if (tid == 0) {
        int n = 0;
        for (int i = 0; i < lim; ++i) {
          if (vsup[start + i] == 0) {
            s_kbox[n] = box[start + i];
            s_klab[n] = lab[start + i];
            ++n;
          }
        }
        s_nkept = n;
      }
    }
    __syncthreads();

    // Phase 2: all threads apply kept chunk boxes to all later candidates.
    const int nk = s_nkept;
    if (nk > 0) {
      for (int j = start + 32 + tid; j < KTOT; j += bs) {
        if (sup[j]) continue;
        const float4 bj = box[j];
        const int lj = lab[j];
        const float areaj = (bj.z - bj.x) * (bj.w - bj.y);
        for (int t = 0; t < nk; ++t) {
          if (s_klab[t] != lj) continue;
          const float4 bi = s_kbox[t];
          const float iw = fminf(bi.z, bj.z) - fmaxf(bi.x, bj.x);
          const float ih = fminf(bi.w, bj.w) - fmaxf(bi.y, bj.y);
          const float inter = fmaxf(iw, 0.0f) * fmaxf(ih, 0.0f);
          const float area_i = (bi.z - bi.x) * (bi.w - bi.y);
          const float uni = fmaxf(area_i + areaj - inter, 1e-9f);
          if (inter > NMS_THR * uni) { sup[j] = 1; break; }
        }
      }
    }
    __syncthreads();
  }

  // Compact the first DET kept detections (300 rows; thread 0).
  if (tid == 0) {
    int cnt = 0;
    for (int k = 0; k < KTOT && cnt < DET; ++k) {
      if (!sup[k]) {
        const float4 bk = box[k];
        float* o = out + ((size_t)b * DET + cnt) * 6;
        o[0] = bk.x; o[1] = bk.y; o[2] = bk.z; o[3] = bk.w;
        o[4] = sort_score[bK + k];
        o[5] = (float)lab[k];
        ++cnt;
      }
    }
  }
}

// ---------------------------------------------------------------------------
// Host-side launch. d_in: feat0..feat4 [B,N,84] f32, image_shapes [B,2] i32.
// d_out: [B, 300, 6] f32.
// ---------------------------------------------------------------------------
extern "C" void kernel_launch(void* const* d_in, const int* in_sizes, int n_in,
                              void* d_out, int out_size, void* d_ws, size_t ws_size,
                              hipStream_t stream) {
  const float* f0 = (const float*)d_in[0];
  const float* f1 = (const float*)d_in[1];
  const float* f2 = (const float*)d_in[2];
  const float* f3 = (const float*)d_in[3];
  const float* f4 = (const float*)d_in[4];
  const int* image_shapes = (const int*)d_in[5];
  float* out = (float*)d_out;

  // Workspace layout (bytes), 16-aligned.
  char* ws = (char*)d_ws;
  const size_t nBK = (size_t)BATCH * KTOT;          // 20000
  float* cand_score = (float*)(ws + 0);             // nBK f32
  int*   cand_label = (int*)(ws + nBK * 4);
  float* cand_box   = (float*)(ws + nBK * 8);       // nBK float4
  float* sort_score = (float*)(ws + nBK * 24);
  int*   sort_label = (int*)(ws + nBK * 28);
  float* sort_box   = (float*)(ws + nBK * 32);      // nBK float4
  unsigned* hist    = (unsigned*)(ws + nBK * 48);   // B*5*1024 u32
  char* tail        = ws + nBK * 48 + (size_t)BATCH * NUM_LVL * 1024 * 4;
  int* cutv    = (int*)(tail + 0);
  int* budgetv = (int*)(tail + 128);
  int* sel_cnt = (int*)(tail + 256);
  int* cut_cnt = (int*)(tail + 384);
  // total ~1.05 MB

  // Zero the global histograms (re-zeroed every call for graph replay).
  hipMemsetAsync(hist, 0, (size_t)BATCH * NUM_LVL * 1024 * 4, stream);

  // Stage 1a: device-wide histogram.
  hist_kernel<<<dim3(CHUNKS_MAX, NUM_LVL, BATCH), 256, 0, stream>>>(
      f0, f1, f2, f3, f4, hist);

  // Stage 1b: cutoff + candidate padding init.
  cutoff_kernel<<<80, 256, 0, stream>>>(
      hist, cutv, budgetv, sel_cnt, cut_cnt, cand_score, cand_label, cand_box);

  // Stage 1c: slot claim + decode.
  select_decode_kernel<<<dim3(CHUNKS_MAX, NUM_LVL, BATCH), 256, 0, stream>>>(
      f0, f1, f2, f3, f4, image_shapes, cutv, budgetv, sel_cnt, cut_cnt,
      cand_score, cand_label, cand_box);

  // Stage 2: per-image descending sort (64 KB dynamic LDS).
  sort_candidates_kernel<<<BATCH, 1024, 8192 * sizeof(unsigned long long), stream>>>(
      cand_score, cand_label, cand_box, sort_score, sort_label, sort_box);

  // Stage 3: chunked NMS + compaction (K*16 + K*4 + K bytes LDS, padded).
  const size_t nms_lds = (size_t)KTOT * 16 + (size_t)KTOT * 4 + (size_t)KTOT + 16;
  nms_compact_kernel<<<BATCH, 1024, nms_lds, stream>>>(
      sort_score, sort_label, sort_box, out);
}